// RayMarcher_10625749090423
// MI455X (gfx1250) — compile-verified
//
#include <hip/hip_runtime.h>
#include <cstdint>
#include <cstddef>

// RayMarcher for MI455X (gfx1250, wave32).
// Memory-bound (~0.66 flop/byte, ~11.5us HBM floor @ 23.3 TB/s):
//  - rgb tile (1536 contiguous bytes per ray) is DMA'd into LDS by the
//    CDNA5 Tensor Data Mover (tensor_load_to_lds, TENSORcnt), overlapping
//    the transmittance scan on densities. Fallback: async-to-LDS b128 path.
//  - one wave per ray; lane owns 4 contiguous samples; shfl product scan.

#define WAVES_PER_BLOCK 8
#define PTS 128
#define RAY_RGB_BYTES (PTS * 3 * 4)   // 1536 B per ray = 192 x 8B elements

typedef unsigned int u32x4 __attribute__((ext_vector_type(4)));
typedef int          i32x4 __attribute__((ext_vector_type(4)));
typedef int          i32x8 __attribute__((ext_vector_type(8)));

#if defined(__HIP_DEVICE_COMPILE__) && __has_builtin(__builtin_amdgcn_tensor_load_to_lds)
#define USE_TDM 1
#else
#define USE_TDM 0
#endif

// therock-10.0 headers (amdgpu-toolchain) => 6-arg builtin; ROCm 7.2 => 5-arg
#if __has_include(<hip/amd_detail/amd_gfx1250_TDM.h>)
#define TDM_ARITY6 1
#else
#define TDM_ARITY6 0
#endif

__global__ __launch_bounds__(WAVES_PER_BLOCK * 32)
void raymarch_kernel(const float* __restrict__ dens,
                     const float* __restrict__ rgbp,
                     float* __restrict__ out,
                     int nRays)
{
    __shared__ __align__(16) float tile[WAVES_PER_BLOCK * PTS * 3]; // 12 KB

    const int lane = threadIdx.x & 31;
    const int wave = threadIdx.x >> 5;
    const int ray  = blockIdx.x * WAVES_PER_BLOCK + wave;
    if (ray >= nRays) return;                      // wave-uniform guard

    // ---- stage 1: launch DMA of this ray's rgb (1536 B) into LDS ----
    const char* gsrc = (const char*)rgbp + (size_t)ray * RAY_RGB_BYTES;
    // Flat pointer to LDS: addr[31:0] is the LDS byte offset (ISA 10.2).
    const unsigned ldsBase =
        (unsigned)(uintptr_t)(const void*)&tile[wave * PTS * 3];

#if USE_TDM
    {
        // Tensor DMA descriptor (D#), 1-D tile: 192 x 8B = 1536 B.
        const unsigned glo = (unsigned)(uintptr_t)gsrc;
        const unsigned ghi = (unsigned)((uintptr_t)gsrc >> 32);
        u32x4 g0;
        g0.x = 1u;                                             // count=1, user D#
        g0.y = (unsigned)__builtin_amdgcn_readfirstlane((int)ldsBase); // lds_addr
        g0.z = (unsigned)__builtin_amdgcn_readfirstlane((int)glo);     // gaddr lo
        g0.w = ((unsigned)__builtin_amdgcn_readfirstlane((int)ghi) & 0x01FFFFFFu)
               | 0x80000000u;                                  // gaddr hi | type=2
        i32x8 g1;
        g1[0] = 0x00030000;            // wg_mask=0, data_size=3 (8 B elements)
        g1[1] = (int)(192u << 16);     // tensor_dim0 = 192 (lo16), abar=0
        g1[2] = 0x00010000;            // tensor_dim0 hi=0, tensor_dim1=1
        g1[3] = (int)(192u << 16);     // tile_dim0 = 192, tensor_dim1 hi=0
        g1[4] = 0;                     // tile_dim1 = tile_dim2 = 0 (1-D tile)
        g1[5] = 192;                   // tensor_dim0_stride = 192
        g1[6] = 0;                     // stride hi, tensor_dim1_stride lo
        g1[7] = 0;                     // tensor_dim1_stride hi
        i32x4 gz = {0, 0, 0, 0};
#if TDM_ARITY6
        i32x8 gz8 = {0, 0, 0, 0, 0, 0, 0, 0};
        __builtin_amdgcn_tensor_load_to_lds(g0, g1, gz, gz, gz8, 0);
#else
        __builtin_amdgcn_tensor_load_to_lds(g0, g1, gz, gz, 0);
#endif
    }
#else
    // Fallback: 96 x 16B chunks = 3 async b128 ops per lane (ASYNCcnt).
#pragma unroll
    for (int i = 0; i < 3; ++i) {
        const char* g = gsrc + (i * 32 + lane) * 16;
        unsigned    l = ldsBase + (unsigned)((i * 32 + lane) * 16);
        asm volatile("global_load_async_to_lds_b128 %0, %1, off"
                     :: "v"(l), "v"(g) : "memory");
    }
#endif

    // ---- stage 2: densities + transmittance scan (overlaps the DMA) ----
    const float4 d4 =
        *reinterpret_cast<const float4*>(dens + (size_t)ray * PTS + lane * 4);

    // f = 1 + EPS - d  (1e-10 is below f32 ulp at 1.0, same as XLA's f32 fold)
    const float f0 = 1.0f - d4.x;
    const float f1 = 1.0f - d4.y;
    const float f2 = 1.0f - d4.z;
    const float f3 = 1.0f - d4.w;
    const float cp = f0 * f1 * f2 * f3;            // chunk product

    // inclusive product scan across 32 lanes (5 steps)
    float inc = cp;
#pragma unroll
    for (int s = 1; s < 32; s <<= 1) {
        float t = __shfl_up(inc, (unsigned)s, 32);
        inc = (lane >= s) ? inc * t : inc;
    }
    // exclusive prefix for this lane's chunk
    float excl = __shfl_up(inc, 1u, 32);
    excl = (lane == 0) ? 1.0f : excl;

    const float T0 = excl;
    const float T1 = T0 * f0;
    const float T2 = T1 * f1;
    const float T3 = T2 * f2;
    const float w0 = d4.x * T0;
    const float w1 = d4.y * T1;
    const float w2 = d4.z * T2;
    const float w3 = d4.w * T3;

    // opacity partial: product of (1 - d) over this lane's 4 samples
    float g = f0 * f1 * f2 * f3;

    // depth partial: linspace(0.1, 6.0, 128)
    const float step = (6.0f - 0.1f) / 127.0f;
    const float p0   = (float)(lane * 4);
    float dep = w0 * (0.1f + p0 * step)
              + w1 * (0.1f + (p0 + 1.0f) * step)
              + w2 * (0.1f + (p0 + 2.0f) * step)
              + w3 * (0.1f + (p0 + 3.0f) * step);

    // ---- stage 3: wait for DMA, consume rgb tile from LDS ----
#if USE_TDM
#if __has_builtin(__builtin_amdgcn_s_wait_tensorcnt)
    __builtin_amdgcn_s_wait_tensorcnt(0);
#else
    asm volatile("s_wait_tensorcnt 0x0" ::: "memory");
#endif
    asm volatile("" ::: "memory");   // order LDS reads after the wait
#else
    asm volatile("s_wait_asynccnt 0x0" ::: "memory");
#endif

    const float* t = &tile[wave * PTS * 3 + lane * 12];        // 48B-aligned
    const float4 c0 = *reinterpret_cast<const float4*>(t + 0); // p0.rgb p1.r
    const float4 c1 = *reinterpret_cast<const float4*>(t + 4); // p1.gb p2.rg
    const float4 c2 = *reinterpret_cast<const float4*>(t + 8); // p2.b p3.rgb

    float cr = w0 * c0.x + w1 * c0.w + w2 * c1.z + w3 * c2.y;
    float cg = w0 * c0.y + w1 * c1.x + w2 * c1.w + w3 * c2.z;
    float cb = w0 * c0.z + w1 * c1.y + w2 * c2.x + w3 * c2.w;

    // ---- stage 4: wave reductions (sum / product butterflies) ----
#pragma unroll
    for (int s = 16; s > 0; s >>= 1) {
        cr  += __shfl_xor(cr,  s, 32);
        cg  += __shfl_xor(cg,  s, 32);
        cb  += __shfl_xor(cb,  s, 32);
        dep += __shfl_xor(dep, s, 32);
        g   *= __shfl_xor(g,   s, 32);
    }

    if (lane == 0) {
        const size_t N = (size_t)nRays;
        out[(size_t)ray * 3 + 0] = cr;             // rgb_calc [B,R,3]
        out[(size_t)ray * 3 + 1] = cg;
        out[(size_t)ray * 3 + 2] = cb;
        out[N * 3 + (size_t)ray] = 1.0f - g;       // opacity  [B,R,1]
        out[N * 4 + (size_t)ray] = dep;            // depth    [B,R]
    }
}

extern "C" void kernel_launch(void* const* d_in, const int* in_sizes, int n_in,
                              void* d_out, int out_size, void* d_ws, size_t ws_size,
                              hipStream_t stream) {
    (void)n_in; (void)out_size; (void)d_ws; (void)ws_size;
    const float* dens = (const float*)d_in[0];   // [B,R,P,1] f32
    const float* rgb  = (const float*)d_in[1];   // [B,R,P,3] f32
    float* out        = (float*)d_out;           // [3N | N | N] f32

    const int nRays  = in_sizes[0] / PTS;        // B*R = 131072
    const int blocks = (nRays + WAVES_PER_BLOCK - 1) / WAVES_PER_BLOCK;

    raymarch_kernel<<<dim3(blocks), dim3(WAVES_PER_BLOCK * 32), 0, stream>>>(
        dens, rgb, out, nRays);
}